// SymbolsEncoder_6210522710683
// MI455X (gfx1250) — compile-verified
//
#include <hip/hip_runtime.h>
#include <math.h>

#define DIM 128
#define PSTR 144    // float2 stride per K-pair row of W in LDS (bank-split halves)
#define X_STR 260   // 256 + 4 pad (floats)

typedef float v2f __attribute__((ext_vector_type(2)));
typedef float v8f __attribute__((ext_vector_type(8)));

// Order-preserving mapping float -> uint so atomicMax(uint) == float max.
__device__ __forceinline__ unsigned f2ord(float f) {
    unsigned u = __float_as_uint(f);
    return (u & 0x80000000u) ? ~u : (u | 0x80000000u);
}
__device__ __forceinline__ float ord2f(unsigned u) {
    return (u & 0x80000000u) ? __uint_as_float(u & 0x7fffffffu)
                             : __uint_as_float(~u);
}

// ---------------- init: zero agg/denom, segmax = -inf (ordered 0) ----------
__global__ void k_init(float* __restrict__ agg, unsigned* __restrict__ segmax,
                       float* __restrict__ denom, int S) {
    long long i = (long long)blockIdx.x * blockDim.x + threadIdx.x;
    if (i < (long long)S * DIM) agg[i] = 0.0f;
    if (i < S) { segmax[i] = 0u; denom[i] = 0.0f; }
}

// ---------------- scores + segment max: one wave32 per occurrence ----------
__global__ void k_score(const float* __restrict__ enc_id,
                        const float* __restrict__ enc_ast,
                        const int*   __restrict__ sym_idx,
                        const int*   __restrict__ occ_node,
                        const int*   __restrict__ occ_sym,
                        float*       __restrict__ scores,
                        unsigned*    __restrict__ segmax,
                        int E) {
    int e    = (int)((blockIdx.x * blockDim.x + threadIdx.x) >> 5);
    int lane = threadIdx.x & 31;
    if (e >= E) return;
    int s = occ_sym[e];
    long long node = occ_node[e];
    long long id   = sym_idx[s];
    float4 o = *(const float4*)(enc_ast + node * DIM + lane * 4);
    float4 q = *(const float4*)(enc_id  + id   * DIM + lane * 4);
    float p = o.x * q.x + o.y * q.y + o.z * q.z + o.w * q.w;
    #pragma unroll
    for (int off = 16; off > 0; off >>= 1) p += __shfl_xor(p, off, 32);
    float sc = p * 0.08838834764831845f;   // 1/sqrt(128)
    if (lane == 0) {
        scores[e] = sc;
        atomicMax(&segmax[s], f2ord(sc));
    }
}

// ------------- exp weights + segment sum + weighted scatter-add ------------
__global__ void k_accum(const float* __restrict__ enc_ast,
                        const int*   __restrict__ occ_node,
                        const int*   __restrict__ occ_sym,
                        const float* __restrict__ scores,
                        const unsigned* __restrict__ segmax,
                        float* __restrict__ denom,
                        float* __restrict__ agg,
                        int E) {
    int e    = (int)((blockIdx.x * blockDim.x + threadIdx.x) >> 5);
    int lane = threadIdx.x & 31;
    if (e >= E) return;
    int s = occ_sym[e];
    long long node = occ_node[e];
    float w = __expf(scores[e] - ord2f(segmax[s]));
    if (lane == 0) atomicAdd(&denom[s], w);
    float4 o = *(const float4*)(enc_ast + node * DIM + lane * 4);
    float* arow = agg + (long long)s * DIM + lane * 4;
    atomicAdd(arow + 0, w * o.x);
    atomicAdd(arow + 1, w * o.y);
    atomicAdd(arow + 2, w * o.z);
    atomicAdd(arow + 3, w * o.w);
}

// ---------------- normalize agg by denom (0 for empty segments) ------------
__global__ void k_norm(float* __restrict__ agg, const float* __restrict__ denom,
                       int S) {
    long long i = (long long)blockIdx.x * blockDim.x + threadIdx.x;
    if (i >= (long long)S * DIM) return;
    float d = denom[i >> 7];
    agg[i] = (d > 0.0f) ? agg[i] / d : 0.0f;
}

// --------- fused concat + GEMM + ReLU via V_WMMA_F32_16X16X4_F32 -----------
// Block: 256 threads = 8 waves, 128 symbol rows per block; each wave owns a
// 16(M) x 128(N) output strip. W staged K-pair-interleaved so each WMMA B
// fragment is ONE aligned ds_load_b64 into an even VGPR pair; A/B fragments
// double-buffered so DS loads overlap the WMMA bursts. Epilogue: branch-free
// immediate-offset stores for interior blocks, per-row guard only in the
// single boundary block.
__global__ __launch_bounds__(256) void k_gemm(
        const float* __restrict__ enc_id,
        const int*   __restrict__ sym_idx,
        const float* __restrict__ Wc,
        const float* __restrict__ agg,
        float*       __restrict__ out,
        int S) {
    extern __shared__ float smem[];
    float* Wp = smem;                        // [128 K-pairs][PSTR] float2s
    float* Xs = smem + 128 * PSTR * 2;       // [128][X_STR] floats
    int tid  = threadIdx.x;
    int row0 = blockIdx.x * 128;

    // Stage W_combiner [2D][D] as K-pair-interleaved float2s:
    //   Wp[k>>1][c] = ( W[k][c], W[k+1][c] )
    for (int i = tid; i < 256 * DIM; i += 256) {
        int k = i >> 7, c = i & (DIM - 1);
        Wp[((k >> 1) * PSTR + c) * 2 + (k & 1)] = Wc[i];
    }

    // Stage X tile: X[r] = [ q_row | agg_row ], q gathered on the fly.
    for (int i = tid; i < 128 * DIM; i += 256) {
        int r = i >> 7, c = i & (DIM - 1);
        int s = row0 + r; if (s >= S) s = S - 1;       // clamp (stores guarded)
        long long id = sym_idx[s];
        Xs[r * X_STR + c]       = enc_id[id * DIM + c];
        Xs[r * X_STR + DIM + c] = agg[(long long)s * DIM + c];
    }
    __syncthreads();

    int lane  = tid & 31;
    int wv    = tid >> 5;          // wave id 0..7
    int l16   = lane & 15;
    int half  = lane >> 4;         // 0: K 0-1, 1: K 2-3  (16x4 f32 A layout)
    int mbase = wv * 16;

    v8f acc[8] = {};
    const float* xrow = Xs + (mbase + l16) * X_STR + half * 2;
    // B fragment base: pair index (k/2 + half), column l16.
    const v2f* wrow = (const v2f*)(Wp + (half * PSTR + l16) * 2);

    // Double-buffered fragments: prefetch k+4 while computing k.
    v2f abuf[2];
    v2f bbuf[2][8];
    abuf[0] = *(const v2f*)(xrow);
    #pragma unroll
    for (int ct = 0; ct < 8; ++ct) bbuf[0][ct] = wrow[ct * 16];

    #pragma unroll 2
    for (int k = 0; k < 2 * DIM; k += 4) {
        int cur = (k >> 2) & 1, nxt = cur ^ 1;
        // Unconditional prefetch of step k+4. The one-past-the-end tail read
        // (k==252) stays inside this workgroup's LDS allocation (Xs row pad /
        // Xs region), loads junk that is never consumed.
        int kn = k + 4;
        abuf[nxt] = *(const v2f*)(xrow + kn);
        const v2f* wk = wrow + (kn >> 1) * PSTR;
        #pragma unroll
        for (int ct = 0; ct < 8; ++ct) bbuf[nxt][ct] = wk[ct * 16];

        #pragma unroll
        for (int ct = 0; ct < 8; ++ct)
            acc[ct] = __builtin_amdgcn_wmma_f32_16x16x4_f32(
                false, abuf[cur], false, bbuf[cur][ct], (short)0, acc[ct],
                false, false);
    }

    // C/D layout: VGPR v -> M = v (lanes 0-15) or v+8 (lanes 16-31), N = l16.
    int rlane = row0 + mbase + (half ? 8 : 0);       // this lane's first row
    float* obase = out + ((long long)rlane * DIM + l16);
    if (row0 + 128 <= S) {
        // Interior block: branch-free, immediate-offset stores.
        #pragma unroll
        for (int v = 0; v < 8; ++v)
            #pragma unroll
            for (int ct = 0; ct < 8; ++ct)
                obase[v * DIM + ct * 16] = fmaxf(acc[ct][v], 0.0f);
    } else {
        // Boundary block: one guard per output row.
        #pragma unroll
        for (int v = 0; v < 8; ++v) {
            if (rlane + v < S) {
                #pragma unroll
                for (int ct = 0; ct < 8; ++ct)
                    obase[v * DIM + ct * 16] = fmaxf(acc[ct][v], 0.0f);
            }
        }
    }
}

extern "C" void kernel_launch(void* const* d_in, const int* in_sizes, int n_in,
                              void* d_out, int out_size, void* d_ws, size_t ws_size,
                              hipStream_t stream) {
    const float* enc_id  = (const float*)d_in[0];
    const float* enc_ast = (const float*)d_in[1];
    const float* Wc      = (const float*)d_in[2];
    const int*   sym_idx = (const int*)d_in[3];
    const int*   occ_node= (const int*)d_in[4];
    const int*   occ_sym = (const int*)d_in[5];
    int S = in_sizes[3];
    int E = in_sizes[4];
    float* out = (float*)d_out;

    // Workspace layout (floats): agg[S*128] | scores[E] | denom[S] | segmax[S]
    float*    agg    = (float*)d_ws;
    float*    scores = agg + (size_t)S * DIM;
    float*    denom  = scores + E;
    unsigned* segmax = (unsigned*)(denom + S);

    const int threads = 256;
    long long nSD = (long long)S * DIM;
    int gInit = (int)((nSD + threads - 1) / threads);
    k_init<<<gInit, threads, 0, stream>>>(agg, segmax, denom, S);

    int wavesPerBlock = threads / 32;
    int gOcc = (E + wavesPerBlock - 1) / wavesPerBlock;
    k_score<<<gOcc, threads, 0, stream>>>(enc_id, enc_ast, sym_idx,
                                          occ_node, occ_sym, scores, segmax, E);
    k_accum<<<gOcc, threads, 0, stream>>>(enc_ast, occ_node, occ_sym,
                                          scores, segmax, denom, agg, E);
    k_norm<<<gInit, threads, 0, stream>>>(agg, denom, S);

    int gGemm = (S + 127) / 128;
    size_t lds = (size_t)(128 * PSTR * 2 + 128 * X_STR) * sizeof(float); // ~281 KB
    k_gemm<<<gGemm, threads, lds, stream>>>(enc_id, sym_idx, Wc, agg, out, S);
}